// CustomPooling_80401787781600
// MI455X (gfx1250) — compile-verified
//
#include <hip/hip_runtime.h>

typedef __attribute__((ext_vector_type(2))) float v2f;
typedef __attribute__((ext_vector_type(8))) float v8f;

#define LDIM 13
#define BDIM 32
#define SDIM 512
#define HDIM 768
#define LAYER0 9            // layers -4..-1 of 13
#define SCHUNK 128
#define NPARTS 16           // 4 layers * 4 s-chunks

// workspace layout in floats
#define WS_MASK 0                       // [B][2][S]
#define WS_CNT  (BDIM * 2 * SDIM)       // [B][2]
#define WS_PART (WS_CNT + BDIM * 2)     // [B][NPARTS][2][H]

// ---------------------------------------------------------------------------
// Kernel 1: build term/text masks + counts.  One block per batch, thread = s.
// ---------------------------------------------------------------------------
__global__ void mask_kernel(const int* __restrict__ ids,
                            const int* __restrict__ attn,
                            const int* __restrict__ tt,
                            float* __restrict__ ws) {
    __shared__ int red[SDIM];
    const int b = blockIdx.x;
    const int s = threadIdx.x;

    const int id = ids[b * SDIM + s];
    // first PAD position (emulates the Python break / cumprod)
    red[s] = (id == 0) ? s : SDIM;
    __syncthreads();
    for (int st = SDIM / 2; st > 0; st >>= 1) {
        if (s < st) red[s] = min(red[s], red[s + st]);
        __syncthreads();
    }
    const int firstPad = red[0];
    __syncthreads();

    const bool valid = (s < firstPad) && (id != 101) && (id != 102) &&
                       (attn[b * SDIM + s] == 1);
    const int ttv = tt[b * SDIM + s];
    const float term = (valid && ttv == 0) ? 1.f : 0.f;
    const float text = (valid && ttv == 1) ? 1.f : 0.f;
    ws[WS_MASK + (b * 2 + 0) * SDIM + s] = term;
    ws[WS_MASK + (b * 2 + 1) * SDIM + s] = text;

    // packed count reduction (counts <= 512, fit in 16 bits each)
    red[s] = (int)term + ((int)text << 16);
    __syncthreads();
    for (int st = SDIM / 2; st > 0; st >>= 1) {
        if (s < st) red[s] += red[s + st];
        __syncthreads();
    }
    if (s == 0) {
        const int packed = red[0];
        ws[WS_CNT + b * 2 + 0] = (float)(packed & 0xFFFF);
        ws[WS_CNT + b * 2 + 1] = (float)(packed >> 16);
    }
}

// ---------------------------------------------------------------------------
// Kernel 2: WMMA masked-sum partials.
//   grid.x = b * NPARTS + part, part = layer_idx*4 + s_chunk
//   256 threads = 8 waves; wave w accumulates h-tiles (w + 8*j)*16, j=0..5.
//   D = A x B + C with A = hs^T tile (M=h, K=s), B = mask (K=s, N=mask type).
// ---------------------------------------------------------------------------
__global__ void wmma_partial_kernel(const float* __restrict__ hs,
                                    float* __restrict__ ws) {
    __shared__ float lmask[2 * SCHUNK];

    const int bp    = blockIdx.x;
    const int b     = bp / NPARTS;
    const int part  = bp % NPARTS;
    const int layer = LAYER0 + (part >> 2);
    const int sbase = (part & 3) * SCHUNK;
    const int tid   = threadIdx.x;

    // stage this chunk's masks in LDS (exactly 256 floats, 256 threads)
    {
        const int nn = tid / SCHUNK;
        const int ss = tid % SCHUNK;
        lmask[tid] = ws[WS_MASK + (b * 2 + nn) * SDIM + sbase + ss];
    }
    __syncthreads();

    const int w    = tid >> 5;        // wave id 0..7
    const int lane = tid & 31;
    const int n    = lane & 15;       // N column (mask type when < 2), also M row for A
    const int kh   = lane >> 4;       // K half: lanes 0-15 hold K=0,1; 16-31 hold K=2,3

    v8f c[6];
    const v8f zero = {0.f, 0.f, 0.f, 0.f, 0.f, 0.f, 0.f, 0.f};
#pragma unroll
    for (int j = 0; j < 6; ++j) c[j] = zero;

    const float* hb = hs + (((size_t)layer * BDIM + b) * SDIM) * HDIM;
    const int nc = (n < 2) ? n : 0;   // safe LDS index; zeroed below when n>=2

#pragma unroll 2
    for (int s0 = 0; s0 < SCHUNK; s0 += 4) {
        // B operand: B[k, n] = mask_n[sbase + s0 + k]; zero for n >= 2
        v2f bv;
        const float mx = lmask[nc * SCHUNK + s0 + 2 * kh];
        const float my = lmask[nc * SCHUNK + s0 + 2 * kh + 1];
        bv.x = (n < 2) ? mx : 0.f;
        bv.y = (n < 2) ? my : 0.f;

        // A operand rows for K = 2*kh and 2*kh+1 (consecutive s, same h)
        const float* arow = hb + (size_t)(sbase + s0 + 2 * kh) * HDIM;

#pragma unroll
        for (int j = 0; j < 6; ++j) {
            const int h = (w + 8 * j) * 16 + n;   // M = lane&15
            v2f av;
            av.x = arow[h];            // 16 lanes -> 64B contiguous segment
            av.y = arow[HDIM + h];
            c[j] = __builtin_amdgcn_wmma_f32_16x16x4_f32(
                false, av, false, bv, (short)0, c[j], false, false);
        }
    }

    // D layout: vgpr r, lanes 0-15 -> D[M=r, N=lane]; lanes 16-31 -> D[M=r+8, N=lane-16]
    // M = h_local, N = mask type. Lanes with n<2 hold 8 consecutive-h sums each.
    if (n < 2) {
#pragma unroll
        for (int j = 0; j < 6; ++j) {
            const int hbm = (w + 8 * j) * 16 + kh * 8;
            float* dst = ws + WS_PART +
                         (((size_t)b * NPARTS + part) * 2 + n) * HDIM + hbm;
#pragma unroll
            for (int r = 0; r < 8; ++r) dst[r] = c[j][r];
        }
    }
}

// ---------------------------------------------------------------------------
// Kernel 3: reduce 16 partials, apply layer-mean (0.25) and divide by counts.
// ---------------------------------------------------------------------------
__global__ void finalize_kernel(const float* __restrict__ ws,
                                float* __restrict__ out) {
    const int t = blockIdx.x * blockDim.x + threadIdx.x;   // 0 .. B*2*H-1
    const int b   = t / (2 * HDIM);
    const int rem = t % (2 * HDIM);
    const int n   = rem / HDIM;
    const int h   = rem % HDIM;

    float sum = 0.f;
#pragma unroll
    for (int p = 0; p < NPARTS; ++p)
        sum += ws[WS_PART + (((size_t)b * NPARTS + p) * 2 + n) * HDIM + h];

    const float cnt = ws[WS_CNT + b * 2 + n];
    out[t] = sum * 0.25f / cnt;
}

// ---------------------------------------------------------------------------
extern "C" void kernel_launch(void* const* d_in, const int* in_sizes, int n_in,
                              void* d_out, int out_size, void* d_ws, size_t ws_size,
                              hipStream_t stream) {
    const float* hs   = (const float*)d_in[0];  // [13,32,512,768] f32
    const int*   ids  = (const int*)d_in[1];    // [32,512]
    const int*   attn = (const int*)d_in[2];    // [32,512]
    const int*   tt   = (const int*)d_in[3];    // [32,512]
    float* ws  = (float*)d_ws;
    float* out = (float*)d_out;                 // [32, 1536] f32

    mask_kernel<<<BDIM, SDIM, 0, stream>>>(ids, attn, tt, ws);
    wmma_partial_kernel<<<BDIM * NPARTS, 256, 0, stream>>>(hs, ws);
    finalize_kernel<<<(BDIM * 2 * HDIM) / 256, 256, 0, stream>>>(ws, out);
}